// DMCU_DAE_35347580846308
// MI455X (gfx1250) — compile-verified
//
#include <hip/hip_runtime.h>
#include <hip/hip_bf16.h>
#include <math.h>

#define N_NODES 32768
#define N_EDGES 524288
#define IN_F    256
#define OUT_F   128
#define NB      4

typedef __attribute__((ext_vector_type(2))) float v2f;
typedef __attribute__((ext_vector_type(8))) float v8f;

// ---------------------------------------------------------------------------
// Softmax over the 4 branch gates (single thread; trivial cost).
// ---------------------------------------------------------------------------
__global__ void gates_kernel(const float* __restrict__ g,
                             const float* __restrict__ temp,
                             float* __restrict__ out) {
    if (threadIdx.x == 0 && blockIdx.x == 0) {
        float t = *temp;
        float v[NB];
        float m = -1e30f;
        for (int i = 0; i < NB; ++i) { v[i] = g[i] / t; m = fmaxf(m, v[i]); }
        float s = 0.f;
        for (int i = 0; i < NB; ++i) { v[i] = __expf(v[i] - m); s += v[i]; }
        for (int i = 0; i < NB; ++i) out[i] = v[i] / s;
    }
}

// ---------------------------------------------------------------------------
// Elementwise helpers
// ---------------------------------------------------------------------------
__global__ void zero_kernel(float* __restrict__ p, int n) {
    int i = blockIdx.x * blockDim.x + threadIdx.x;
    if (i < n) p[i] = 0.f;
}

__global__ void init_out_kernel(float* __restrict__ out,
                                const float* __restrict__ bo, int n) {
    int i = blockIdx.x * blockDim.x + threadIdx.x;
    if (i < n) out[i] = bo[i & (OUT_F - 1)];
}

__global__ void deg_kernel(const long long* __restrict__ edges,
                           float* __restrict__ deg, int nEdges) {
    int e = blockIdx.x * blockDim.x + threadIdx.x;
    if (e < nEdges) {
        long long c = edges[(size_t)nEdges + e];   // col = edge_index[1]
        atomicAdd(&deg[c], 1.0f);
    }
}

__global__ void invdeg_kernel(float* __restrict__ deg, int n) {
    int i = blockIdx.x * blockDim.x + threadIdx.x;
    if (i < n) deg[i] = 1.0f / fmaxf(deg[i], 1.0f);
}

// h[n][f] *= inv_deg[n]
__global__ void scale_rows_kernel(float* __restrict__ h,
                                  const float* __restrict__ invdeg, int n) {
    int i = blockIdx.x * blockDim.x + threadIdx.x;
    if (i < n) h[i] *= invdeg[i >> 7];   // OUT_F == 128
}

// ---------------------------------------------------------------------------
// Scatter-add: one wave32 per edge; each lane handles 4 consecutive features.
// hout[col[e]][:] += hin[row[e]][:]
// ---------------------------------------------------------------------------
__global__ __launch_bounds__(256)
void scatter_add_kernel(const float* __restrict__ hin,
                        float* __restrict__ hout,
                        const long long* __restrict__ edges, int nEdges) {
    int gwave = (blockIdx.x * blockDim.x + threadIdx.x) >> 5;
    int lane  = threadIdx.x & 31;
    if (gwave >= nEdges) return;
    long long src = edges[gwave];                    // row = edge_index[0]
    long long dst = edges[(size_t)nEdges + gwave];   // col = edge_index[1]
    const float4 v = *(const float4*)(hin + (size_t)src * OUT_F + lane * 4);
    float* o = hout + (size_t)dst * OUT_F + lane * 4;
    atomicAdd(o + 0, v.x);
    atomicAdd(o + 1, v.y);
    atomicAdd(o + 2, v.z);
    atomicAdd(o + 3, v.w);
}

// ---------------------------------------------------------------------------
// fp32 WMMA GEMM:  C[M x 128] = act(A[M x K] @ W[K x 128] + bias) * scale
// 256 threads = 8 waves; block tile 128x128; K staged through LDS in 32-chunks.
// Wave w owns rows [w*16, w*16+16) x all 128 cols = 8 accumulator tiles.
//
// V_WMMA_F32_16X16X4_F32 fragment layouts (ISA 7.12.2):
//   A 16x4 : lanes 0-15 -> (M=lane, K=kk..kk+1), lanes 16-31 -> (M=lane-16, K=kk+2..kk+3)
//   B 4x16 : VGPR v: lanes 0-15 -> (K=kk+v, N=lane), lanes 16-31 -> (K=kk+2+v, N=lane-16)
//   C 16x16: VGPR v: lanes 0-15 -> (M=v, N=lane),    lanes 16-31 -> (M=v+8, N=lane-16)
//
// A is staged with GLOBAL_LOAD_ASYNC_TO_LDS_B128 (GVS: SGPR base + per-lane
// 32-bit offset); B is staged through VGPRs because its LDS layout is a
// k-pair transpose-interleave (Bs2[j][n] = {B[2j][n], B[2j+1][n]}) so each
// B fragment is a single aligned ds_load_b64 into an even VGPR pair.
// ---------------------------------------------------------------------------
#define KC      32
#define AS_STR  36    // floats; 8B frag loads hit all 64 banks exactly once
#define BP_STR  144   // float2 pairs; 2*144 dwords == 32 mod 64 -> halves disjoint

__global__ __launch_bounds__(256)
void gemm_f32_wmma(const float* __restrict__ A, const float* __restrict__ W,
                   const float* __restrict__ bias, float* __restrict__ C,
                   int M, int K, int relu, int acc,
                   const float* __restrict__ scalep) {
    __shared__ float As[128 * AS_STR];
    __shared__ v2f   Bs2[(KC / 2) * BP_STR];

    const int tid  = threadIdx.x;
    const int wave = tid >> 5;
    const int lane = tid & 31;
    const int l15  = lane & 15;
    const int hi   = lane >> 4;            // 0 for lanes 0-15, 1 for 16-31
    const int rowBase = blockIdx.x * 128;

    const unsigned ldsABase = (unsigned)(size_t)&As[0];

    v8f cacc[8];
#pragma unroll
    for (int t = 0; t < 8; ++t)
#pragma unroll
        for (int j = 0; j < 8; ++j) cacc[t][j] = 0.f;

    for (int k0 = 0; k0 < K; k0 += KC) {
        // ---- Stage A chunk (128 x 32 f32) asynchronously into LDS --------
        // Each thread issues 4 x b128 async copies (contiguous on both sides).
#pragma unroll
        for (int p = 0; p < 4; ++p) {
            int r = (tid >> 3) + p * 32;
            int c = (tid & 7) * 4;
            unsigned ldsOff = ldsABase + (unsigned)((r * AS_STR + c) * 4);
            unsigned gOff   = (unsigned)(((size_t)(rowBase + r) * K + k0 + c) * 4);
            asm volatile("global_load_async_to_lds_b128 %0, %1, %2"
                         :: "v"(ldsOff), "v"(gOff), "s"(A) : "memory");
        }

        // ---- Stage B chunk (32 x 128 f32) as k-pair-interleaved float2 ---
        // Thread -> pair-row pr (16 of them) x 8-column group.
        {
            int pr = tid >> 4;            // 0..15
            int c  = (tid & 15) * 8;      // 0..120
            const float* w0 = W + (size_t)(k0 + 2 * pr) * OUT_F + c;
            const float* w1 = W + (size_t)(k0 + 2 * pr + 1) * OUT_F + c;
            float4 r0a = *(const float4*)(w0);
            float4 r0b = *(const float4*)(w0 + 4);
            float4 r1a = *(const float4*)(w1);
            float4 r1b = *(const float4*)(w1 + 4);
            v2f* dst = &Bs2[pr * BP_STR + c];
            dst[0] = (v2f){r0a.x, r1a.x};
            dst[1] = (v2f){r0a.y, r1a.y};
            dst[2] = (v2f){r0a.z, r1a.z};
            dst[3] = (v2f){r0a.w, r1a.w};
            dst[4] = (v2f){r0b.x, r1b.x};
            dst[5] = (v2f){r0b.y, r1b.y};
            dst[6] = (v2f){r0b.z, r1b.z};
            dst[7] = (v2f){r0b.w, r1b.w};
        }

        // Fence this wave's async copies, then barrier with the others.
        asm volatile("s_wait_asynccnt 0" ::: "memory");
        __syncthreads();

        const int arow = wave * 16 + l15;
#pragma unroll
        for (int kk = 0; kk < KC; kk += 4) {
            const int kl = kk + hi * 2;
            const int jp = (kk >> 1) + hi;               // B pair-row
            v2f afrag = *(const v2f*)&As[arow * AS_STR + kl];
#pragma unroll
            for (int t = 0; t < 8; ++t) {
                v2f bfrag = Bs2[jp * BP_STR + t * 16 + l15];
                cacc[t] = __builtin_amdgcn_wmma_f32_16x16x4_f32(
                    false, afrag, false, bfrag, (short)0, cacc[t],
                    false, false);
            }
        }
        __syncthreads();
    }

    // Epilogue: bias -> relu -> gate scale -> (accumulate|store)
    const float scale = scalep ? *scalep : 1.0f;
#pragma unroll
    for (int t = 0; t < 8; ++t) {
        const int col = t * 16 + l15;
        const float b = bias ? bias[col] : 0.f;
#pragma unroll
        for (int v = 0; v < 8; ++v) {
            int row = rowBase + wave * 16 + v + hi * 8;
            float val = cacc[t][v] + b;
            if (relu) val = fmaxf(val, 0.f);
            val *= scale;
            size_t idx = (size_t)row * OUT_F + col;
            if (acc) C[idx] += val;
            else     C[idx] = val;
        }
    }
}

// ---------------------------------------------------------------------------
// Host-side orchestration
// ---------------------------------------------------------------------------
extern "C" void kernel_launch(void* const* d_in, const int* in_sizes, int n_in,
                              void* d_out, int out_size, void* d_ws, size_t ws_size,
                              hipStream_t stream) {
    (void)in_sizes; (void)n_in; (void)out_size; (void)ws_size;

    const float*     x     = (const float*)d_in[0];
    const long long* edges = (const long long*)d_in[1];
    const float*     Wb    = (const float*)d_in[2];
    const float*     bb    = (const float*)d_in[3];
    const float*     W1    = (const float*)d_in[4];
    const float*     b1    = (const float*)d_in[5];
    const float*     W2    = (const float*)d_in[6];
    const float*     b2    = (const float*)d_in[7];
    const float*     bg    = (const float*)d_in[8];
    const float*     temp  = (const float*)d_in[9];
    const float*     Wo    = (const float*)d_in[10];
    const float*     bo    = (const float*)d_in[11];
    float*           out   = (float*)d_out;

    // Workspace layout
    char*  ws    = (char*)d_ws;
    float* gates = (float*)ws;                               // 4 floats (pad 256B)
    float* deg   = (float*)(ws + 256);                       // N floats -> inv_deg
    float* bufA  = (float*)(ws + 256 + (size_t)N_NODES * 4); // N x 128
    float* bufB  = bufA + (size_t)N_NODES * OUT_F;           // N x 128

    const int NH = N_NODES * OUT_F;       // 4M elements

    // gates = softmax(branch_gates / temperature)
    gates_kernel<<<1, 32, 0, stream>>>(bg, temp, gates);

    // degree -> inv_deg (in place)
    zero_kernel<<<(N_NODES + 255) / 256, 256, 0, stream>>>(deg, N_NODES);
    deg_kernel<<<(N_EDGES + 255) / 256, 256, 0, stream>>>(edges, deg, N_EDGES);
    invdeg_kernel<<<(N_NODES + 255) / 256, 256, 0, stream>>>(deg, N_NODES);

    // out = bo (broadcast), then accumulate gated branch GEMMs into it
    init_out_kernel<<<(NH + 255) / 256, 256, 0, stream>>>(out, bo, NH);

    const int gemmGrid = N_NODES / 128;   // 256 blocks

    for (int i = 0; i < NB; ++i) {
        // h = x @ Wb[i] + bb[i]
        gemm_f32_wmma<<<gemmGrid, 256, 0, stream>>>(
            x, Wb + (size_t)i * IN_F * OUT_F, bb + i * OUT_F,
            bufA, N_NODES, IN_F, 0, 0, nullptr);

        // i scatter-mean propagation steps
        for (int s = 0; s < i; ++s) {
            zero_kernel<<<(NH + 255) / 256, 256, 0, stream>>>(bufB, NH);
            scatter_add_kernel<<<(N_EDGES * 32) / 256, 256, 0, stream>>>(
                bufA, bufB, edges, N_EDGES);
            scale_rows_kernel<<<(NH + 255) / 256, 256, 0, stream>>>(bufB, deg, NH);
            float* t = bufA; bufA = bufB; bufB = t;
        }

        // t = relu(h @ W1[i] + b1[i])
        gemm_f32_wmma<<<gemmGrid, 256, 0, stream>>>(
            bufA, W1 + (size_t)i * OUT_F * OUT_F, b1 + i * OUT_F,
            bufB, N_NODES, OUT_F, 1, 0, nullptr);

        // h2 = t @ W2[i] + b2[i]
        gemm_f32_wmma<<<gemmGrid, 256, 0, stream>>>(
            bufB, W2 + (size_t)i * OUT_F * OUT_F, b2 + i * OUT_F,
            bufA, N_NODES, OUT_F, 0, 0, nullptr);

        // out += gates[i] * (h2 @ Wo[i*128 : (i+1)*128, :])
        gemm_f32_wmma<<<gemmGrid, 256, 0, stream>>>(
            bufA, Wo + (size_t)i * OUT_F * OUT_F, nullptr,
            out, N_NODES, OUT_F, 0, 1, gates + i);
    }
}